// SpectralMultiheadAttention_40321152974851
// MI455X (gfx1250) — compile-verified
//
#include <hip/hip_runtime.h>
#include <math.h>
#include <stdint.h>

// ---------------- problem constants ----------------
#define BATCH   4
#define SEQ     4096
#define DIM     1024
#define HEADS   16
#define HDIM    64
#define FBINS   2049                  // SEQ/2 + 1
#define OUTSZ   (HEADS * FBINS * 2)   // 65568
#define LN_EPS  1e-5f

typedef __attribute__((ext_vector_type(2))) float v2f;
typedef __attribute__((ext_vector_type(8))) float v8f;
typedef __attribute__((ext_vector_type(4))) unsigned int v4u;
typedef __attribute__((ext_vector_type(4))) int v4i;
typedef __attribute__((ext_vector_type(8))) int v8i;

// ---------------- gfx1250 feature probes ----------------
#if defined(__has_builtin)
#if __has_builtin(__builtin_amdgcn_tensor_load_to_lds_d2)
#define SMA_HAS_TDM 1
#define SMA_TDM_D2 1
#elif __has_builtin(__builtin_amdgcn_tensor_load_to_lds)
#define SMA_HAS_TDM 1
#endif
#if __has_builtin(__builtin_amdgcn_global_load_async_to_lds_b32)
#define SMA_HAS_ASYNC_LD 1
#endif
#if __has_builtin(__builtin_amdgcn_global_store_async_from_lds_b32)
#define SMA_HAS_ASYNC_ST 1
#endif
#endif

// async b32 builtins take typed AS-qualified int pointers (per hipcc diag):
typedef __attribute__((address_space(1))) int as1_int;
typedef __attribute__((address_space(3))) int as3_int;
#define SMA_AS1(p) ((as1_int*)(uintptr_t)(p))
#define SMA_AS3(p) ((as3_int*)(unsigned)(uintptr_t)(p))

__device__ __forceinline__ void sma_wait_async0() {
#if __has_builtin(__builtin_amdgcn_s_wait_asynccnt)
    __builtin_amdgcn_s_wait_asynccnt(0);
#else
    asm volatile("s_wait_asynccnt 0x0" ::: "memory");
#endif
}

__device__ __forceinline__ void sma_wait_tensor0() {
#if __has_builtin(__builtin_amdgcn_s_wait_tensorcnt)
    __builtin_amdgcn_s_wait_tensorcnt(0);
#else
    asm volatile("s_wait_tensorcnt 0x0" ::: "memory");
#endif
}

// ============================================================
// Kernel 1: LayerNorm. One block (256 thr) per (b, n) row.
// ============================================================
__global__ void sma_ln_kernel(const float* __restrict__ x,
                              const float* __restrict__ ln_w,
                              const float* __restrict__ ln_b,
                              float* __restrict__ xn) {
    const int row = blockIdx.x;                   // b*SEQ + n
    const float* xr = x + (size_t)row * DIM;
    float*      orr = xn + (size_t)row * DIM;
    __shared__ float rs[256];
    __shared__ float rs2[256];
    float v[4];
    float s = 0.f, s2 = 0.f;
#pragma unroll
    for (int i = 0; i < 4; ++i) {
        v[i] = xr[threadIdx.x + 256 * i];
        s  += v[i];
        s2 += v[i] * v[i];
    }
    rs[threadIdx.x] = s;  rs2[threadIdx.x] = s2;
    __syncthreads();
    for (int off = 128; off > 0; off >>= 1) {
        if (threadIdx.x < off) {
            rs[threadIdx.x]  += rs[threadIdx.x + off];
            rs2[threadIdx.x] += rs2[threadIdx.x + off];
        }
        __syncthreads();
    }
    const float mean = rs[0] * (1.f / DIM);
    const float var  = rs2[0] * (1.f / DIM) - mean * mean;
    const float inv  = rsqrtf(var + LN_EPS);
#pragma unroll
    for (int i = 0; i < 4; ++i) {
        const int c = threadIdx.x + 256 * i;
        orr[c] = (v[i] - mean) * inv * ln_w[c] + ln_b[c];
    }
}

// ============================================================
// Kernel 2: tiled transpose per batch: [rows][cols] -> [cols][rows]
// ============================================================
__global__ void sma_transpose_kernel(const float* __restrict__ in,
                                     float* __restrict__ out,
                                     int rows, int cols) {
    __shared__ float tile[32][33];
    const int b = blockIdx.z;
    const float* ib = in  + (size_t)b * rows * cols;
    float*       ob = out + (size_t)b * rows * cols;
    const int c0 = blockIdx.x * 32, r0 = blockIdx.y * 32;
#pragma unroll
    for (int i = 0; i < 32; i += 8)
        tile[threadIdx.y + i][threadIdx.x] =
            ib[(size_t)(r0 + threadIdx.y + i) * cols + (c0 + threadIdx.x)];
    __syncthreads();
#pragma unroll
    for (int i = 0; i < 32; i += 8)
        ob[(size_t)(c0 + threadIdx.y + i) * rows + (r0 + threadIdx.x)] =
            tile[threadIdx.x][threadIdx.y + i];
}

// ============================================================
// Kernel 3: context mean into 16-row padded tile. grid (DIM, 16)
// ============================================================
__global__ void sma_ctx_kernel(const float* __restrict__ xnt,
                               float* __restrict__ ctx16) {
    const int c = blockIdx.x;
    const int b = blockIdx.y;
    if (b >= BATCH) {
        if (threadIdx.x == 0) ctx16[(size_t)b * DIM + c] = 0.f;
        return;
    }
    const float* p = xnt + ((size_t)b * DIM + c) * SEQ;
    float s = 0.f;
    for (int n = threadIdx.x; n < SEQ; n += 256) s += p[n];
    __shared__ float rs[256];
    rs[threadIdx.x] = s;
    __syncthreads();
    for (int off = 128; off > 0; off >>= 1) {
        if (threadIdx.x < off) rs[threadIdx.x] += rs[threadIdx.x + off];
        __syncthreads();
    }
    if (threadIdx.x == 0) ctx16[(size_t)b * DIM + c] = rs[0] * (1.f / SEQ);
}

// ============================================================
// Kernel 4: 16xN = 16xK @ KxN (+bias, opt relu) via f32 WMMA.
// One wave per 16x16 tile; block = 4 waves.
// ============================================================
__global__ void sma_gemm16_wmma(const float* __restrict__ A,
                                const float* __restrict__ W,
                                const float* __restrict__ bias,
                                float* __restrict__ D,
                                int K, int N, int do_relu, int ntiles) {
    const int wave = threadIdx.x >> 5;
    const int tile = blockIdx.x * 4 + wave;
    if (tile >= ntiles) return;                        // whole-wave uniform
    const int n0   = tile * 16;
    const int lane = threadIdx.x & 31;
    const int half = lane >> 4;
    const int l15  = lane & 15;

    v8f acc = {};
    for (int k = 0; k < K; k += 4) {
        const int ka = k + half * 2;
        v2f a;
        a.x = A[(size_t)l15 * K + ka];
        a.y = A[(size_t)l15 * K + ka + 1];
        v2f bm;
        bm.x = W[(size_t)ka * N + n0 + l15];
        bm.y = W[(size_t)(ka + 1) * N + n0 + l15];
        if (k + 4 < K)
            __builtin_prefetch(&W[(size_t)(k + 4) * N + n0 + l15], 0, 1);
        acc = __builtin_amdgcn_wmma_f32_16x16x4_f32(
                  false, a, false, bm, (short)0, acc, false, false);
    }
    const int col   = n0 + l15;
    const float bv  = bias[col];
    const int rbase = half * 8;
#pragma unroll
    for (int i = 0; i < 8; ++i) {
        float v = acc[i] + bv;
        if (do_relu) v = fmaxf(v, 0.f);
        D[(size_t)(rbase + i) * N + col] = v;
    }
}

// ============================================================
// Kernel 5: in-LDS 4096-pt FFT + spectral modulation + iFFT.
// Input row DMA'd into LDS (TDM or async-to-LDS); output row
// DMA'd out via async store-from-LDS. Guarded fallbacks.
// ============================================================
__device__ __forceinline__ float2 cmulf(float2 a, float2 b) {
    return make_float2(a.x * b.x - a.y * b.y, a.x * b.y + a.y * b.x);
}

__device__ __forceinline__ void sma_fft_stages(float2* buf, const float2* tw,
                                               int tid, bool inverse) {
#pragma unroll 1
    for (int s = 1; s <= 12; ++s) {
        const int half  = 1 << (s - 1);
        const int shift = 12 - s;
        for (int j = tid; j < 2048; j += 256) {
            const int p   = j & (half - 1);
            const int idx = ((j >> (s - 1)) << s) + p;
            float2 w = tw[p << shift];          // e^{-2pi i q/4096}
            if (inverse) w.y = -w.y;
            const float2 u = buf[idx];
            const float2 t = cmulf(w, buf[idx + half]);
            buf[idx]        = make_float2(u.x + t.x, u.y + t.y);
            buf[idx + half] = make_float2(u.x - t.x, u.y - t.y);
        }
        __syncthreads();
    }
}

union SmaTwStage {                 // twiddles & I/O staging overlay (16 KB)
    float2 tw[2048];
    float  stage[4096];
};

__global__ void sma_fft_mod_kernel(const float* __restrict__ xnt,
                                   const float* __restrict__ base_filter,
                                   const float* __restrict__ base_bias,
                                   const float* __restrict__ adapt,
                                   float* __restrict__ yt) {
    const int bc = blockIdx.x;            // b*DIM + c
    const int b  = bc >> 10;
    const int c  = bc & (DIM - 1);
    const int h  = c >> 6;                // head
    const int tid = threadIdx.x;

    __shared__ float2 buf[4096];          // 32 KB
    __shared__ SmaTwStage twu;            // 16 KB

    const float* xin = xnt + (size_t)bc * SEQ;

    // ---- A) input row -> LDS staging ----
#if defined(SMA_HAS_TDM)
    if (tid < 32) {                        // one wave issues the DMA
        const unsigned lds_off = (unsigned)(uintptr_t)(&twu.stage[0]);
        const unsigned long long ga = (unsigned long long)(uintptr_t)xin;
        v4u g0;
        g0.x = 1u;                                          // count=1
        g0.y = lds_off;                                     // lds_addr
        g0.z = (unsigned)(ga & 0xFFFFFFFFu);                // global_addr lo
        g0.w = (unsigned)((ga >> 32) & 0x1FFFFFFu) | (2u << 30); // hi | type=2
        v8i g1;
        g1[0] = (int)(2u << 16);        // data_size = 4B
        g1[1] = (int)(4096u << 16);     // tensor_dim0 = 4096
        g1[2] = (int)(1u << 16);        // tensor_dim1 = 1
        g1[3] = (int)(4096u << 16);     // tile_dim0 = 4096
        g1[4] = 1;                      // tile_dim1 = 1
        g1[5] = 4096;                   // tensor_dim0_stride
        g1[6] = 0;
        g1[7] = 0;
#if defined(SMA_TDM_D2)
        __builtin_amdgcn_tensor_load_to_lds_d2(g0, g1, 0);
#else
        v4i z4 = {0, 0, 0, 0};          // groups 2/3: tile_dim3/4 = 0 (unused)
        v8i z8 = {0, 0, 0, 0, 0, 0, 0, 0};
        __builtin_amdgcn_tensor_load_to_lds(g0, g1, z4, z4, z8, 0);
#endif
        sma_wait_tensor0();
    }
#elif defined(SMA_HAS_ASYNC_LD)
#pragma unroll
    for (int j = 0; j < 16; ++j) {
        const int n = tid + 256 * j;
        __builtin_amdgcn_global_load_async_to_lds_b32(
            SMA_AS1(xin + n), SMA_AS3(&twu.stage[n]), 0, 0);
    }
    sma_wait_async0();
#else
#pragma unroll
    for (int j = 0; j < 16; ++j) {
        const int n = tid + 256 * j;
        twu.stage[n] = xin[n];
    }
#endif
    __syncthreads();

    // ---- B) bit-reversed scatter into complex buffer ----
#pragma unroll
    for (int j = 0; j < 16; ++j) {
        const int n = tid + 256 * j;
        buf[__brev((unsigned)n) >> 20] = make_float2(twu.stage[n], 0.f);
    }
    __syncthreads();

    // ---- C) twiddles (overwrite staging): tw[q] = e^{-2pi i q/4096} ----
    for (int q = tid; q < 2048; q += 256) {
        float sn, cs;
        sincosf(-6.283185307179586f * (float)q * (1.f / 4096.f), &sn, &cs);
        twu.tw[q] = make_float2(cs, sn);
    }
    __syncthreads();

    sma_fft_stages(buf, twu.tw, tid, /*inverse=*/false);

    // ---- D) spectral modulation + magnitude-GELU activation ----
    float2 G[9];
#pragma unroll
    for (int j = 0; j < 9; ++j) {
        const int k = tid + 256 * j;
        if (k <= 2048) {
            const float2 F = buf[k];
            const size_t ai = ((size_t)b * (HEADS * FBINS) + (size_t)h * FBINS + k) * 2;
            const float sc = adapt[ai + 0];
            const float bi = adapt[ai + 1];
            const float ef = base_filter[h * FBINS + k] * (1.f + sc);
            const float eb = base_bias[h * FBINS + k] + bi;
            float2 g = make_float2(F.x * ef + eb, F.y * ef);
            const float mag = sqrtf(g.x * g.x + g.y * g.y);
            const float t   = mag + 0.044715f * mag * mag * mag;
            const float gel = 0.5f * mag * (1.f + tanhf(0.7978845608028654f * t));
            const float s   = gel / (mag + 1e-6f);
            G[j] = make_float2(g.x * s, g.y * s);
        }
    }
    __syncthreads();
    // hermitian-extended spectrum, written in bit-reversed order
#pragma unroll
    for (int j = 0; j < 9; ++j) {
        const int k = tid + 256 * j;
        if (k <= 2048) {
            buf[__brev((unsigned)k) >> 20] = G[j];
            if (k >= 1 && k <= 2047)
                buf[__brev((unsigned)(4096 - k)) >> 20] = make_float2(G[j].x, -G[j].y);
        }
    }
    __syncthreads();

    sma_fft_stages(buf, twu.tw, tid, /*inverse=*/true);

    // ---- E) real part * 1/N -> LDS staging -> async DMA out ----
    float* yout = yt + (size_t)bc * SEQ;
    const float invN = 1.f / 4096.f;
#if defined(SMA_HAS_ASYNC_ST)
#pragma unroll
    for (int j = 0; j < 16; ++j) {
        const int n = tid + 256 * j;
        twu.stage[n] = buf[n].x * invN;
    }
    __syncthreads();
#pragma unroll
    for (int j = 0; j < 16; ++j) {
        const int n = tid + 256 * j;
        __builtin_amdgcn_global_store_async_from_lds_b32(
            SMA_AS1(yout + n), SMA_AS3(&twu.stage[n]), 0, 0);
    }
    sma_wait_async0();
#else
#pragma unroll
    for (int j = 0; j < 16; ++j) {
        const int n = tid + 256 * j;
        yout[n] = buf[n].x * invN;
    }
#endif
}

// ============================================================
// launch
// ============================================================
extern "C" void kernel_launch(void* const* d_in, const int* in_sizes, int n_in,
                              void* d_out, int out_size, void* d_ws, size_t ws_size,
                              hipStream_t stream) {
    const float* x           = (const float*)d_in[0];
    const float* ln_w        = (const float*)d_in[1];
    const float* ln_b        = (const float*)d_in[2];
    const float* base_filter = (const float*)d_in[3];
    const float* base_bias   = (const float*)d_in[4];
    const float* w0          = (const float*)d_in[5];
    const float* b0          = (const float*)d_in[6];
    const float* w1          = (const float*)d_in[7];
    const float* b1          = (const float*)d_in[8];
    const float* w2          = (const float*)d_in[9];
    const float* b2          = (const float*)d_in[10];
    float* out = (float*)d_out;

    const size_t big = (size_t)BATCH * SEQ * DIM;     // 16M floats
    float* xn    = (float*)d_ws;                      // [b][n][c]; reused as yt
    float* xnt   = xn + big;                          // [b][c][n]
    float* ctx16 = xnt + big;                         // 16 x 1024
    float* h0    = ctx16 + 16 * DIM;
    float* h1    = h0 + 16 * DIM;
    float* adapt = h1 + 16 * DIM;                     // 16 x 65568

    sma_ln_kernel<<<BATCH * SEQ, 256, 0, stream>>>(x, ln_w, ln_b, xn);
    sma_transpose_kernel<<<dim3(DIM / 32, SEQ / 32, BATCH), dim3(32, 8), 0, stream>>>(
        xn, xnt, SEQ, DIM);
    sma_ctx_kernel<<<dim3(DIM, 16), 256, 0, stream>>>(xnt, ctx16);
    {
        const int t0 = DIM / 16;
        sma_gemm16_wmma<<<(t0 + 3) / 4, 128, 0, stream>>>(ctx16, w0, b0, h0, DIM, DIM, 1, t0);
        sma_gemm16_wmma<<<(t0 + 3) / 4, 128, 0, stream>>>(h0, w1, b1, h1, DIM, DIM, 1, t0);
        const int t2 = OUTSZ / 16;                    // 4098
        sma_gemm16_wmma<<<(t2 + 3) / 4, 128, 0, stream>>>(h1, w2, b2, adapt, DIM, OUTSZ, 0, t2);
    }
    sma_fft_mod_kernel<<<BATCH * DIM, 256, 0, stream>>>(
        xnt, base_filter, base_bias, adapt, xn);
    sma_transpose_kernel<<<dim3(SEQ / 32, DIM / 32, BATCH), dim3(32, 8), 0, stream>>>(
        xn, out, DIM, SEQ);
}